// NormalDecoder_12489764897254
// MI455X (gfx1250) — compile-verified
//
#include <hip/hip_runtime.h>
#include <hip/hip_bf16.h>

typedef float v2f __attribute__((ext_vector_type(2)));
typedef float v4f __attribute__((ext_vector_type(4)));
typedef float v8f __attribute__((ext_vector_type(8)));

// D = A(16x4) * B(4x16) + C(16x16), all fp32, wave32.
// args: (neg_a, A, neg_b, B, c_mod, C, reuse_a, reuse_b)
#define WMMA_F32(a, b, c) \
    __builtin_amdgcn_wmma_f32_16x16x4_f32(false, (a), false, (b), (short)0, (c), false, false)

__global__ __launch_bounds__(256)
void normal_decoder_kernel(const float* __restrict__ F,   // [N,128]
                           const float* __restrict__ W1,  // [32,128]
                           const float* __restrict__ g1,  // [32]
                           const float* __restrict__ b1,  // [32]
                           const float* __restrict__ W2,  // [16,32]
                           const float* __restrict__ g2,  // [16]
                           const float* __restrict__ b2,  // [16]
                           const float* __restrict__ W3,  // [3,16]
                           const float* __restrict__ b3,  // [3]
                           float* __restrict__ out)       // [N,3]
{
    // act1: 16 rows x 32 cols, stride 34 (even pad -> b64-aligned frag loads, no bank conflicts)
    __shared__ float act1[8][16 * 34];
    // act2: 16 rows x 16 cols, stride 20 (keeps 16B alignment for b128 row reads)
    __shared__ float act2[8][16 * 20];

    const int tid  = threadIdx.x;
    const int wave = tid >> 5;
    const int lane = tid & 31;
    const int n    = lane & 15;   // column / row-in-tile index
    const int kh   = lane >> 4;   // K-half selector for A/B frags; row-half for C
    const int rowbase = blockIdx.x * 128 + wave * 16;

    // ---------------- GEMM1: [16,128] x W1^T -> [16,32] ----------------
    // A frag: lane holds row (n), K = 4*kc + 2*kh + {0,1}
    const float* Ap  = F  + (size_t)(rowbase + n) * 128 + 2 * kh;
    const float* B0p = W1 + (size_t)n        * 128 + 2 * kh;  // out cols 0..15
    const float* B1p = W1 + (size_t)(n + 16) * 128 + 2 * kh;  // out cols 16..31

    v8f c0 = {};
    v8f c1 = {};
#pragma unroll
    for (int kc = 0; kc < 32; ++kc) {
        v2f a  = __builtin_nontemporal_load((const v2f*)(Ap + 4 * kc)); // stream, don't wash L2
        v2f bb0 = *(const v2f*)(B0p + 4 * kc);                          // L2-resident weights
        v2f bb1 = *(const v2f*)(B1p + 4 * kc);
        c0 = WMMA_F32(a, bb0, c0);
        c1 = WMMA_F32(a, bb1, c1);
    }

    // ---------------- LayerNorm(32) + leaky, C-layout -> LDS (A-layout staging) ----
    const float gA = g1[n], bA = b1[n], gB = g1[n + 16], bB = b1[n + 16];
#pragma unroll
    for (int j = 0; j < 8; ++j) {
        float v0 = c0[j];
        float v1 = c1[j];
        float s = v0 + v1;            // partial row sum (per lane: 2 of 32 values)
        float q = v0 * v0 + v1 * v1;  // partial row sum of squares
        // reduce across the 16 lanes of this half (rows 0-7 in lanes 0-15, rows 8-15 in 16-31)
#pragma unroll
        for (int m = 8; m >= 1; m >>= 1) {
            s += __shfl_xor(s, m, 16);
            q += __shfl_xor(q, m, 16);
        }
        float mean = s * (1.0f / 32.0f);
        float var  = q * (1.0f / 32.0f) - mean * mean;
        float rstd = rsqrtf(var + 1e-5f);
        float y0 = (v0 - mean) * rstd * gA + bA;
        float y1 = (v1 - mean) * rstd * gB + bB;
        y0 = (y0 >= 0.0f) ? y0 : 0.1f * y0;
        y1 = (y1 >= 0.0f) ? y1 : 0.1f * y1;
        const int M = j + 8 * kh;                 // actual row index in tile
        act1[wave][M * 34 + n]      = y0;         // col n
        act1[wave][M * 34 + 16 + n] = y1;         // col 16+n
    }
    __syncthreads();

    // ---------------- GEMM2: [16,32] x W2^T -> [16,16] ----------------
    const float* A2p = &act1[wave][n * 34 + 2 * kh];
    const float* B2p = W2 + (size_t)n * 32 + 2 * kh;
    v8f d = {};
#pragma unroll
    for (int kc = 0; kc < 8; ++kc) {
        v2f a = *(const v2f*)(A2p + 4 * kc);      // ds_load_b64
        v2f b = *(const v2f*)(B2p + 4 * kc);
        d = WMMA_F32(a, b, d);
    }

    // ---------------- LayerNorm(16) + leaky -> LDS (row-major) ----------------
    const float gC = g2[n], bC = b2[n];
#pragma unroll
    for (int j = 0; j < 8; ++j) {
        float v = d[j];
        float s = v;
        float q = v * v;
#pragma unroll
        for (int m = 8; m >= 1; m >>= 1) {
            s += __shfl_xor(s, m, 16);
            q += __shfl_xor(q, m, 16);
        }
        float mean = s * (1.0f / 16.0f);
        float var  = q * (1.0f / 16.0f) - mean * mean;
        float rstd = rsqrtf(var + 1e-5f);
        float y = (v - mean) * rstd * gC + bC;
        y = (y >= 0.0f) ? y : 0.1f * y;
        act2[wave][(j + 8 * kh) * 20 + n] = y;
    }
    __syncthreads();

    // ---------------- GEMM3 (16 -> 3) + bias + L2 normalize ----------------
    // Every lane computes row r = n (halves duplicate; LDS reads broadcast), kh==0 stores.
    const float* rowp = &act2[wave][n * 20];      // 80-byte stride -> 16B aligned
    v4f xa = *(const v4f*)(rowp + 0);
    v4f xb = *(const v4f*)(rowp + 4);
    v4f xc = *(const v4f*)(rowp + 8);
    v4f xd = *(const v4f*)(rowp + 12);

    float z[3];
#pragma unroll
    for (int c = 0; c < 3; ++c) {
        const float* w = W3 + c * 16;             // 64B offsets, aligned
        v4f wa = *(const v4f*)(w + 0);
        v4f wb = *(const v4f*)(w + 4);
        v4f wc = *(const v4f*)(w + 8);
        v4f wd = *(const v4f*)(w + 12);
        v4f p = xa * wa + xb * wb + xc * wc + xd * wd;
        z[c] = (p[0] + p[1]) + (p[2] + p[3]) + b3[c];
    }
    float nrm = sqrtf(z[0] * z[0] + z[1] * z[1] + z[2] * z[2]);
    float inv = 1.0f / fmaxf(nrm, 1e-12f);

    if (kh == 0) {
        float* o = out + (size_t)(rowbase + n) * 3;
        __builtin_nontemporal_store(z[0] * inv, o + 0);
        __builtin_nontemporal_store(z[1] * inv, o + 1);
        __builtin_nontemporal_store(z[2] * inv, o + 2);
    }
}

extern "C" void kernel_launch(void* const* d_in, const int* in_sizes, int n_in,
                              void* d_out, int out_size, void* d_ws, size_t ws_size,
                              hipStream_t stream) {
    const float* F  = (const float*)d_in[0];
    const float* W1 = (const float*)d_in[1];
    const float* g1 = (const float*)d_in[2];
    const float* b1 = (const float*)d_in[3];
    const float* W2 = (const float*)d_in[4];
    const float* g2 = (const float*)d_in[5];
    const float* b2 = (const float*)d_in[6];
    const float* W3 = (const float*)d_in[7];
    const float* b3 = (const float*)d_in[8];
    float* out = (float*)d_out;

    const int N = in_sizes[0] / 128;   // 1048576
    const int blocks = N / 128;        // 8192 (N divisible by 128 rows/block)
    normal_decoder_kernel<<<blocks, 256, 0, stream>>>(F, W1, g1, b1, W2, g2, b2, W3, b3, out);
}